// MoMEAdaptor_87058987090357
// MI455X (gfx1250) — compile-verified
//
#include <hip/hip_runtime.h>
#include <math.h>

// ---------------------------------------------------------------------------
// Problem constants (from reference): B=2, S=1024, H=2048, D=256, R=32, K=4,
// N_DB=100000.  B*S = 2048 query rows.
// ---------------------------------------------------------------------------
#define BS      2048
#define HID     2048
#define DIM     256
#define RANK    32
#define TOPK    4
#define NDB     100000

#define NW          4          // waves per workgroup in dist kernel
#define QT          16         // queries per tile (WMMA N)
#define NSLICE      25         // DB slices
#define SLICE_ROWS  4000       // NDB / NSLICE
#define SLICE_TILES 250        // SLICE_ROWS / 16
#define QPAD        260        // LDS row pitch (floats) for bank-conflict-free B loads

typedef __attribute__((ext_vector_type(2))) float v2f;
typedef __attribute__((ext_vector_type(8))) float v8f;

// Workspace layout (byte offsets, all 16B aligned)
#define OFF_Q    0u                      // q        [2048,256] f32  (2,097,152)
#define OFF_T    2097152u                // lora mid [2048,32]  f32  (  262,144)
#define OFF_KNH  2359296u                // 0.5*|k|^2 [100000]  f32  (  400,000)
#define OFF_PS   2759296u                // partial scores [2048,25,4] (819,200)
#define OFF_PI   3578496u                // partial idx    [2048,25,4] (819,200)
#define OFF_IDX  4397696u                // final idx [2048,4] i32   (   32,768)
#define OFF_AO   4430464u                // attn out [2048,256] f32  (2,097,152)
#define OFF_T2   6527616u                // lora mid2 [2048,32] f32  (  262,144)

// ---------------------------------------------------------------------------
// Branchless top-4 insert: keeps (s0..s3 / i0..i3) sorted descending with
// pure v_cmp + v_cndmask — no divergent CF, stays in VGPRs (no scratch).
// ---------------------------------------------------------------------------
__device__ static inline void ins4b(float x, int id,
                                    float& s0, float& s1, float& s2, float& s3,
                                    int& i0, int& i1, int& i2, int& i3) {
  const bool g0 = x > s0;
  const bool g1 = x > s1;
  const bool g2 = x > s2;
  const bool g3 = x > s3;
  const float n0 = g0 ? x  : s0;
  const float n1 = g0 ? s0 : (g1 ? x : s1);
  const float n2 = g1 ? s1 : (g2 ? x : s2);
  const float n3 = g2 ? s2 : (g3 ? x : s3);
  const int   m0 = g0 ? id : i0;
  const int   m1 = g0 ? i0 : (g1 ? id : i1);
  const int   m2 = g1 ? i1 : (g2 ? id : i2);
  const int   m3 = g2 ? i2 : (g3 ? id : i3);
  s0 = n0; s1 = n1; s2 = n2; s3 = n3;
  i0 = m0; i1 = m1; i2 = m2; i3 = m3;
}

// SWAPX16 within wave32: xor_mask=0x10 (offset[14:10]), and_mask=0x1f
__device__ static inline float swzf(float v) {
  return __int_as_float(__builtin_amdgcn_ds_swizzle(__float_as_int(v), 0x401F));
}
__device__ static inline int swzi(int v) {
  return __builtin_amdgcn_ds_swizzle(v, 0x401F);
}

// ---------------------------------------------------------------------------
// X[M,Hdim] @ W[Hdim,32] -> T[M,32].  One wave per row (lane = output col).
// ---------------------------------------------------------------------------
__global__ __launch_bounds__(256) void lora_in32(const float* __restrict__ X,
                                                 const float* __restrict__ W,
                                                 float* __restrict__ T,
                                                 int Hdim) {
  const int lane = threadIdx.x & 31;
  const int row  = blockIdx.x * 8 + (threadIdx.x >> 5);
  const float* xr = X + (size_t)row * Hdim;
  float acc = 0.0f;
  for (int h = 0; h < Hdim; h += 4) {
    float x0 = xr[h + 0], x1 = xr[h + 1], x2 = xr[h + 2], x3 = xr[h + 3];
    acc += x0 * W[(h + 0) * RANK + lane];
    acc += x1 * W[(h + 1) * RANK + lane];
    acc += x2 * W[(h + 2) * RANK + lane];
    acc += x3 * W[(h + 3) * RANK + lane];
  }
  T[(size_t)row * RANK + lane] = acc;
}

// ---------------------------------------------------------------------------
// T[M,32] @ W[32,N] -> Y[M,N].  One block per row.
// ---------------------------------------------------------------------------
__global__ __launch_bounds__(256) void lora_out32(const float* __restrict__ T,
                                                  const float* __restrict__ W,
                                                  float* __restrict__ Y,
                                                  int N) {
  __shared__ float tr[RANK];
  const int row = blockIdx.x;
  if (threadIdx.x < RANK) tr[threadIdx.x] = T[(size_t)row * RANK + threadIdx.x];
  __syncthreads();
  for (int col = threadIdx.x; col < N; col += 256) {
    float acc = 0.0f;
#pragma unroll
    for (int r = 0; r < RANK; ++r) acc += tr[r] * W[(size_t)r * N + col];
    Y[(size_t)row * N + col] = acc;
  }
}

// ---------------------------------------------------------------------------
// 0.5 * sum(db_keys[n]^2)  -> knh[n]
// ---------------------------------------------------------------------------
__global__ __launch_bounds__(256) void kn_half(const float* __restrict__ dbk,
                                               float* __restrict__ knh) {
  const int n = blockIdx.x * 256 + threadIdx.x;
  if (n >= NDB) return;
  const float4* r = (const float4*)(dbk + (size_t)n * DIM);
  float s = 0.0f;
#pragma unroll 4
  for (int u = 0; u < DIM / 4; ++u) {
    float4 v = r[u];
    s += v.x * v.x + v.y * v.y + v.z * v.z + v.w * v.w;
  }
  knh[n] = 0.5f * s;
}

// ---------------------------------------------------------------------------
// Distance GEMM + online top-4 per query.
//   A (WMMA SRC0): db tile, M=16 db rows x K=4   -> lane m=l%16, k=(l/16)*2..+1
//   B (WMMA SRC1): q tile,  K=4 x N=16 queries   -> lane n=l%16, k=(l/16)*2..+1
//   C/D: M=db x N=query; lane l holds query l%16 and 8 db rows (+8 if l>=16)
// grid = (128 query tiles, 25 db slices), block = 128 (4 waves)
// ---------------------------------------------------------------------------
__global__ __launch_bounds__(128) void knn_dist_topk(
    const float* __restrict__ q,     // [BS, DIM]
    const float* __restrict__ dbk,   // [NDB, DIM]
    const float* __restrict__ knh,   // [NDB]
    float* __restrict__ pscore,      // [BS, NSLICE, 4]
    int*   __restrict__ pidx)        // [BS, NSLICE, 4]
{
  __shared__ float qs[QT * QPAD];
  __shared__ float cs[NW][QT][TOPK];
  __shared__ int   ci[NW][QT][TOPK];

  const int tid  = threadIdx.x;
  const int wave = tid >> 5;
  const int lane = tid & 31;
  const int qt    = blockIdx.x;
  const int slice = blockIdx.y;
  const int row0  = slice * SLICE_ROWS;

  // stage 16 query rows (reused operand) into LDS, padded pitch 260
  for (int i = tid; i < QT * DIM; i += 128) {
    int r = i >> 8, c = i & (DIM - 1);
    qs[r * QPAD + c] = q[(size_t)(qt * QT + r) * DIM + c];
  }
  __syncthreads();

  const int m16 = lane & 15;
  const int kh  = (lane >> 4) << 1;   // 0 or 2
  const int mo  = (lane >> 4) << 3;   // 0 or 8 (C-tile M offset for this lane)

  float s0 = -1e30f, s1 = -1e30f, s2 = -1e30f, s3 = -1e30f;
  int   i0 = -1, i1 = -1, i2 = -1, i3 = -1;

  for (int t0 = wave * 2; t0 < SLICE_TILES; t0 += NW * 2) {
    const int rbase = row0 + t0 * 16;
    const float* A0 = dbk + (size_t)(rbase + m16) * DIM + kh;
    const float* A1 = A0 + 16 * DIM;
    const float* Bq = qs + m16 * QPAD + kh;

    v8f c0 = {}, c1 = {};
#pragma unroll 8
    for (int kb = 0; kb < DIM; kb += 4) {
      v2f a0 = *(const v2f*)(A0 + kb);
      v2f a1 = *(const v2f*)(A1 + kb);
      v2f b  = *(const v2f*)(Bq + kb);
      c0 = __builtin_amdgcn_wmma_f32_16x16x4_f32(false, a0, false, b,
                                                 (short)0, c0, false, false);
      c1 = __builtin_amdgcn_wmma_f32_16x16x4_f32(false, a1, false, b,
                                                 (short)0, c1, false, false);
    }

    // score = q.k - 0.5|k|^2 for this lane's 16 db rows
    const int base0 = rbase + mo;        // rows for c0 (this lane)
    const int base1 = base0 + 16;        // rows for c1
    const float4 kA0 = *(const float4*)(knh + base0);
    const float4 kA1 = *(const float4*)(knh + base0 + 4);
    const float4 kB0 = *(const float4*)(knh + base1);
    const float4 kB1 = *(const float4*)(knh + base1 + 4);

    float sc[16];
    sc[0]  = c0[0] - kA0.x; sc[1]  = c0[1] - kA0.y;
    sc[2]  = c0[2] - kA0.z; sc[3]  = c0[3] - kA0.w;
    sc[4]  = c0[4] - kA1.x; sc[5]  = c0[5] - kA1.y;
    sc[6]  = c0[6] - kA1.z; sc[7]  = c0[7] - kA1.w;
    sc[8]  = c1[0] - kB0.x; sc[9]  = c1[1] - kB0.y;
    sc[10] = c1[2] - kB0.z; sc[11] = c1[3] - kB0.w;
    sc[12] = c1[4] - kB1.x; sc[13] = c1[5] - kB1.y;
    sc[14] = c1[6] - kB1.z; sc[15] = c1[7] - kB1.w;

    float mx = sc[0];
#pragma unroll
    for (int j = 1; j < 16; ++j) mx = fmaxf(mx, sc[j]);

    // lazy insert: skip whole wave unless some lane beats its current #4
    if (__builtin_amdgcn_ballot_w32(mx > s3) != 0u) {
#pragma unroll
      for (int j = 0; j < 8; ++j)
        ins4b(sc[j], base0 + j, s0, s1, s2, s3, i0, i1, i2, i3);
#pragma unroll
      for (int j = 0; j < 8; ++j)
        ins4b(sc[8 + j], base1 + j, s0, s1, s2, s3, i0, i1, i2, i3);
    }
  }

  // merge lane l with lane l^16 (same query, other M half)
  {
    float os; int oi;
    os = swzf(s0); oi = swzi(i0); ins4b(os, oi, s0, s1, s2, s3, i0, i1, i2, i3);
    os = swzf(s1); oi = swzi(i1); ins4b(os, oi, s0, s1, s2, s3, i0, i1, i2, i3);
    os = swzf(s2); oi = swzi(i2); ins4b(os, oi, s0, s1, s2, s3, i0, i1, i2, i3);
    os = swzf(s3); oi = swzi(i3); ins4b(os, oi, s0, s1, s2, s3, i0, i1, i2, i3);
  }

  if (lane < QT) {
    cs[wave][lane][0] = s0; cs[wave][lane][1] = s1;
    cs[wave][lane][2] = s2; cs[wave][lane][3] = s3;
    ci[wave][lane][0] = i0; ci[wave][lane][1] = i1;
    ci[wave][lane][2] = i2; ci[wave][lane][3] = i3;
  }
  __syncthreads();

  if (tid < QT) {
    float b0 = -1e30f, b1 = -1e30f, b2 = -1e30f, b3 = -1e30f;
    int   j0 = -1, j1 = -1, j2 = -1, j3 = -1;
#pragma unroll
    for (int w = 0; w < NW; ++w)
#pragma unroll
      for (int k = 0; k < TOPK; ++k)
        ins4b(cs[w][tid][k], ci[w][tid][k], b0, b1, b2, b3, j0, j1, j2, j3);
    const size_t base = ((size_t)(qt * QT + tid) * NSLICE + slice) * TOPK;
    pscore[base + 0] = b0; pidx[base + 0] = j0;
    pscore[base + 1] = b1; pidx[base + 1] = j1;
    pscore[base + 2] = b2; pidx[base + 2] = j2;
    pscore[base + 3] = b3; pidx[base + 3] = j3;
  }
}

// ---------------------------------------------------------------------------
// Merge NSLICE partial top-4 lists into final top-4 indices per query
// (ordered ascending distance, matching jax.lax.top_k output order).
// ---------------------------------------------------------------------------
__global__ __launch_bounds__(256) void topk_merge(const float* __restrict__ ps,
                                                  const int* __restrict__ pi,
                                                  int* __restrict__ idxout) {
  const int qg = blockIdx.x * 256 + threadIdx.x;   // 0..2047
  float s0 = -1e30f, s1 = -1e30f, s2 = -1e30f, s3 = -1e30f;
  int   i0 = -1, i1 = -1, i2 = -1, i3 = -1;
  const size_t base = (size_t)qg * NSLICE * TOPK;
  for (int u = 0; u < NSLICE * TOPK; ++u)
    ins4b(ps[base + u], pi[base + u], s0, s1, s2, s3, i0, i1, i2, i3);
  idxout[qg * TOPK + 0] = i0;
  idxout[qg * TOPK + 1] = i1;
  idxout[qg * TOPK + 2] = i2;
  idxout[qg * TOPK + 3] = i3;
}

// ---------------------------------------------------------------------------
// Causal SDPA over gathered kv rows.  One block (256 thr) per query row.
// kv position j (0..S*K-1) is live iff j <= s_i  (top-left aligned tril),
// so at most 1024 kv rows are ever used; db row for position j is idx[b*4096+j].
// ---------------------------------------------------------------------------
__global__ __launch_bounds__(256) void attn_kernel(const float* __restrict__ q,
                                                   const int* __restrict__ idx,
                                                   const float* __restrict__ dbk,
                                                   const float* __restrict__ dbv,
                                                   float* __restrict__ aout) {
  __shared__ __align__(16) float qsh[DIM];
  __shared__ float sc[1024];
  __shared__ int   rows[1024];
  __shared__ float red[256];

  const int tid = threadIdx.x;
  const int i   = blockIdx.x;        // global query row 0..2047
  const int b   = i >> 10;
  const int si  = i & 1023;
  const int nv  = si + 1;

  qsh[tid] = q[(size_t)i * DIM + tid];
  for (int j = tid; j < nv; j += 256) rows[j] = idx[b * 4096 + j];
  __syncthreads();

  // scores (scaled by 1/sqrt(D) = 1/16)
  for (int j = tid; j < nv; j += 256) {
    const float4* kr = (const float4*)(dbk + (size_t)rows[j] * DIM);
    const float4* q4 = (const float4*)qsh;
    float acc = 0.0f;
#pragma unroll 4
    for (int u = 0; u < DIM / 4; ++u) {
      const float4 kv = kr[u];
      const float4 qv = q4[u];
      acc += qv.x * kv.x + qv.y * kv.y + qv.z * kv.z + qv.w * kv.w;
    }
    sc[j] = acc * 0.0625f;
  }
  __syncthreads();

  // block max
  float m = -1e30f;
  for (int j = tid; j < nv; j += 256) m = fmaxf(m, sc[j]);
  red[tid] = m;
  __syncthreads();
  for (int o = 128; o > 0; o >>= 1) {
    if (tid < o) red[tid] = fmaxf(red[tid], red[tid + o]);
    __syncthreads();
  }
  const float gm = red[0];
  __syncthreads();

  // exp + sum
  float ls = 0.0f;
  for (int j = tid; j < nv; j += 256) {
    const float e = __expf(sc[j] - gm);
    sc[j] = e;
    ls += e;
  }
  red[tid] = ls;
  __syncthreads();
  for (int o = 128; o > 0; o >>= 1) {
    if (tid < o) red[tid] += red[tid + o];
    __syncthreads();
  }
  const float inv = 1.0f / red[0];
  __syncthreads();
  for (int j = tid; j < nv; j += 256) sc[j] *= inv;
  __syncthreads();

  // weighted sum of value rows; thread = output dim
  float acc = 0.0f;
#pragma unroll 4
  for (int j = 0; j < nv; ++j)
    acc += sc[j] * dbv[(size_t)rows[j] * DIM + tid];
  aout[(size_t)i * DIM + tid] = acc;
}

// ---------------------------------------------------------------------------
extern "C" void kernel_launch(void* const* d_in, const int* in_sizes, int n_in,
                              void* d_out, int out_size, void* d_ws, size_t ws_size,
                              hipStream_t stream) {
  const float* hs     = (const float*)d_in[0];  // [B,S,H]
  const float* dbk    = (const float*)d_in[1];  // [NDB,D]
  const float* dbv    = (const float*)d_in[2];  // [NDB,D]
  const float* wq_in  = (const float*)d_in[3];  // [H,R]
  const float* wq_out = (const float*)d_in[4];  // [R,D]
  const float* wv_in  = (const float*)d_in[5];  // [D,R]
  const float* wv_out = (const float*)d_in[6];  // [R,H]
  float* out = (float*)d_out;                   // [B,S,H]

  char* ws = (char*)d_ws;
  float* q    = (float*)(ws + OFF_Q);
  float* tmid = (float*)(ws + OFF_T);
  float* knh  = (float*)(ws + OFF_KNH);
  float* ps   = (float*)(ws + OFF_PS);
  int*   pi   = (int*)  (ws + OFF_PI);
  int*   idx  = (int*)  (ws + OFF_IDX);
  float* ao   = (float*)(ws + OFF_AO);
  float* t2   = (float*)(ws + OFF_T2);

  // 1) q = (hs @ wq_in) @ wq_out
  lora_in32<<<BS / 8, 256, 0, stream>>>(hs, wq_in, tmid, HID);
  lora_out32<<<BS, 256, 0, stream>>>(tmid, wq_out, q, DIM);

  // 2) 0.5*|k|^2
  kn_half<<<(NDB + 255) / 256, 256, 0, stream>>>(dbk, knh);

  // 3) WMMA distance GEMM + streaming top-4
  knn_dist_topk<<<dim3(BS / QT, NSLICE), 128, 0, stream>>>(q, dbk, knh, ps, pi);

  // 4) merge slice partials -> final neighbor indices
  topk_merge<<<BS / 256, 256, 0, stream>>>(ps, pi, idx);

  // 5) causal SDPA over gathered kv
  attn_kernel<<<BS, 256, 0, stream>>>(q, idx, dbk, dbv, ao);

  // 6) out = (attn_out @ wv_in) @ wv_out
  lora_in32<<<BS / 8, 256, 0, stream>>>(ao, wv_in, t2, DIM);
  lora_out32<<<BS, 256, 0, stream>>>(t2, wv_out, out, HID);
}